// CleanSelect_29635274342934
// MI455X (gfx1250) — compile-verified
//
#include <hip/hip_runtime.h>

typedef float v2f __attribute__((ext_vector_type(2)));
typedef float v8f __attribute__((ext_vector_type(8)));

#define DIM   2048
#define NINST 64
#define CLEAN 32
#define KC    128
#define LSTR  (KC + 4)   // pad to dodge LDS bank conflicts

// ---------------- Kernel 1: per-split Gram matrix via V_WMMA_F32_16X16X4_F32 ----
// grid = num_split blocks, 512 threads = 16 wave32 -> one 16x16 tile per wave.
__global__ __launch_bounds__(512) void gram_kernel(const float* __restrict__ x,
                                                   float* __restrict__ sim)
{
    const int s = blockIdx.x;
    __shared__ float lds[NINST * LSTR];          // 64 x 128 fp32 panel (~33 KB)
    const float* xs = x + (size_t)s * NINST * DIM;

    const int tid  = threadIdx.x;
    const int lane = tid & 31;                   // wave32
    const int wave = tid >> 5;
    const int ti   = wave >> 2;                  // tile row  (0..3)
    const int tj   = wave & 3;                   // tile col  (0..3)
    const int half = lane >> 4;                  // 0: K=k,k+1   1: K=k+2,k+3
    const int l    = lane & 15;                  // M index (A) / N index (B)

    v8f acc = {0.f, 0.f, 0.f, 0.f, 0.f, 0.f, 0.f, 0.f};

    for (int k0 = 0; k0 < DIM; k0 += KC) {
        __syncthreads();
        // cooperative load 64 x KC floats; 2048 float4s over 512 threads
        for (int i = tid; i < NINST * (KC / 4); i += 512) {
            const int row = i / (KC / 4);
            const int c4  = (i % (KC / 4)) * 4;
            const float4 v = *(const float4*)(xs + (size_t)row * DIM + k0 + c4);
            float* dst = &lds[row * LSTR + c4];
            dst[0] = v.x; dst[1] = v.y; dst[2] = v.z; dst[3] = v.w;
        }
        __syncthreads();

        const float* arow = &lds[(ti * 16 + l) * LSTR];   // A: row M = ti*16+l
        const float* brow = &lds[(tj * 16 + l) * LSTR];   // B = A^T: row N = tj*16+l
        for (int kk = 0; kk < KC; kk += 4) {
            const v2f a = *(const v2f*)(arow + kk + 2 * half);
            const v2f b = *(const v2f*)(brow + kk + 2 * half);
            acc = __builtin_amdgcn_wmma_f32_16x16x4_f32(
                /*neg_a=*/false, a, /*neg_b=*/false, b,
                /*c_mod=*/(short)0, acc, /*reuse_a=*/false, /*reuse_b=*/false);
        }
    }

    // C/D layout: VGPR r -> M = r + 8*half, N = l (within this wave's tile)
    float* simS = sim + (size_t)s * NINST * NINST;
#pragma unroll
    for (int r = 0; r < 8; ++r) {
        const int m = ti * 16 + 8 * half + r;
        const int n = tj * 16 + l;
        simS[m * NINST + n] = acc[r];
    }
}

// ---------------- Kernel 2: ranks -> scores -> stable top-32 selection --------
__global__ __launch_bounds__(64) void select_kernel(const float* __restrict__ sim,
                                                    int* __restrict__ idx)
{
    const int s = blockIdx.x;
    __shared__ float smat[NINST * NINST];
    __shared__ int   sscore[NINST];
    const float* simS = sim + (size_t)s * NINST * NINST;
    const int n = threadIdx.x;

    for (int i = n; i < NINST * NINST; i += NINST) smat[i] = simS[i];
    __syncthreads();

    // stable ascending rank of column n within each row j, summed
    int score = 0;
    for (int j = 0; j < NINST; ++j) {
        const float v = smat[j * NINST + n];
        for (int m = 0; m < NINST; ++m) {
            const float vm = smat[j * NINST + m];
            score += (vm < v) || (vm == v && m < n);
        }
    }
    sscore[n] = score;
    __syncthreads();

    // stable descending rank (argsort(-scores): bigger score first, ties by index)
    int r = 0;
    for (int m = 0; m < NINST; ++m) {
        const int sm = sscore[m];
        r += (sm > score) || (sm == score && m < n);
    }
    if (r < CLEAN) idx[s * CLEAN + r] = n;
}

// ---------------- Kernel 3: gather selected rows ------------------------------
__global__ __launch_bounds__(256) void gather_kernel(const float* __restrict__ x,
                                                     const int* __restrict__ idx,
                                                     float* __restrict__ out)
{
    const int orow = blockIdx.x;                 // 0 .. num_split*CLEAN-1
    const int s    = orow / CLEAN;
    const int src  = idx[orow];
    const float4* sp = (const float4*)(x + ((size_t)s * NINST + src) * DIM);
    float4*       dp = (float4*)(out + (size_t)orow * DIM);
    const int t = threadIdx.x;
    dp[t]       = sp[t];                         // DIM/4 = 512 float4 per row
    dp[t + 256] = sp[t + 256];
}

extern "C" void kernel_launch(void* const* d_in, const int* in_sizes, int n_in,
                              void* d_out, int out_size, void* d_ws, size_t ws_size,
                              hipStream_t stream) {
    (void)n_in; (void)out_size; (void)ws_size;
    const float* x = (const float*)d_in[0];
    const int num_split = in_sizes[0] / (NINST * DIM);   // 16384*2048 / (64*2048) = 256

    float* sim = (float*)d_ws;                                   // 256*64*64*4 = 4 MB
    int*   idx = (int*)((char*)d_ws +
                        (size_t)num_split * NINST * NINST * sizeof(float));
    float* out = (float*)d_out;

    gram_kernel<<<num_split, 512, 0, stream>>>(x, sim);
    select_kernel<<<num_split, NINST, 0, stream>>>(sim, idx);
    gather_kernel<<<num_split * CLEAN, 256, 0, stream>>>(x, idx, out);
}